// embeddingWithWord2Vec_14903536517909
// MI455X (gfx1250) — compile-verified
//
#include <hip/hip_runtime.h>
#include <hip/hip_bf16.h>

// embeddingWithWord2Vec: one_hot(inputs) @ embeddingDict  ==  row gather.
//   inputs:        int32  [128, 512]      (65536 tokens, values in [0, 3882))
//   embeddingDict: float32[3882, 128]     (row = 512 bytes)
//   out:           float32[128, 512, 128] (33.5 MB)
//
// Memory-bound: ~36 MB of HBM traffic -> ~1.5 us at 23.3 TB/s. No FLOPs needed.
//
// Decomposition (identical for both code paths below):
//   1 wave32  = 1 token  (512-byte row == 32 lanes x 16 bytes)
//   1 block   = 256 threads = 8 waves = 8 tokens
//   grid      = 65536 / 8 = 8192 blocks
//
// Primary path: gfx1250 async LDS data mover
//   global_load_async_to_lds_b128  (row -> per-wave LDS slice, ASYNCcnt)
//   s_wait_asynccnt 0              (load->store LDS ordering, per-wave private slice)
//   global_store_async_from_lds_b128 (LDS slice -> contiguous output)
// Fallback: direct b128 load + non-temporal b128 store (keeps the 2 MB
// embedding table resident in L2 while streaming the write-once output past it).

#define EMB_D          128                 // embedding dim (floats)
#define F4_PER_ROW     (EMB_D / 4)         // 32 float4 per row == lanes per wave
#define WAVES_PER_BLK  8
#define BLOCK_THREADS  (WAVES_PER_BLK * 32)

typedef float f4  __attribute__((ext_vector_type(4)));
typedef int   v4i __attribute__((ext_vector_type(4)));

#if defined(__AMDGCN__) && \
    __has_builtin(__builtin_amdgcn_global_load_async_to_lds_b128) && \
    __has_builtin(__builtin_amdgcn_global_store_async_from_lds_b128) && \
    __has_builtin(__builtin_amdgcn_s_wait_asynccnt)
#define USE_GFX1250_ASYNC 1
#else
#define USE_GFX1250_ASYNC 0
#endif

#if USE_GFX1250_ASYNC
// Builtin signature (from clang diagnostic): params are v4i pointers in the
// global (LangAS::cuda_device -> AS1) and shared (AS3) address spaces.
typedef __attribute__((address_space(1))) v4i g_v4i;   // global v4i
typedef __attribute__((address_space(3))) v4i l_v4i;   // LDS v4i
#endif

__global__ __launch_bounds__(BLOCK_THREADS)
void embedding_gather_kernel(const int* __restrict__ idx,
                             const float* __restrict__ emb,
                             float* __restrict__ out,
                             int n_tok)
{
    const int wave = threadIdx.x >> 5;
    const int lane = threadIdx.x & 31;

    int tok = (int)(blockIdx.x * WAVES_PER_BLK + wave);
#if defined(__AMDGCN__)
    // Token id is wave-uniform; force it scalar so idx[tok] becomes an SMEM load
    // and the branch below stays wave-uniform (EXEC all-ones for the async ops).
    tok = __builtin_amdgcn_readfirstlane(tok);
#endif
    if (tok >= n_tok) return;

    const int row = idx[tok];   // uniform per wave

#if USE_GFX1250_ASYNC
    // Per-wave private 512 B LDS staging slice: no cross-wave sharing -> no barriers.
    __shared__ f4 stage[WAVES_PER_BLK * F4_PER_ROW];

    f4* lds_slot = &stage[wave * F4_PER_ROW + lane];

    // Row -> LDS via the async data mover (ASYNCcnt, bypasses VGPRs).
    __builtin_amdgcn_global_load_async_to_lds_b128(
        (g_v4i*)(emb + (size_t)row * EMB_D + (size_t)lane * 4),
        (l_v4i*)lds_slot,
        /*offset=*/0, /*cpol=*/0);

    // Async loads complete in order; wait so the following LDS read (by the
    // async store engine) observes the freshly written row.
    __builtin_amdgcn_s_wait_asynccnt(0);

    // LDS -> contiguous output slab. Trailing completion is covered by the
    // implicit wait-idle at S_ENDPGM.
    __builtin_amdgcn_global_store_async_from_lds_b128(
        (g_v4i*)(out + (size_t)tok * EMB_D + (size_t)lane * 4),
        (l_v4i*)lds_slot,
        /*offset=*/0, /*cpol=*/0);
#else
    // Direct path: b128 load, non-temporal b128 store (output is write-once).
    const f4* src = (const f4*)emb;
    f4 v = src[(size_t)row * F4_PER_ROW + lane];
    __builtin_nontemporal_store(v, (f4*)out + (size_t)tok * F4_PER_ROW + lane);
#endif
}

extern "C" void kernel_launch(void* const* d_in, const int* in_sizes, int n_in,
                              void* d_out, int out_size, void* d_ws, size_t ws_size,
                              hipStream_t stream) {
    const int*   idx = (const int*)  d_in[0];   // [128*512] int32
    const float* emb = (const float*)d_in[1];   // [3882*128] float32
    float*       out = (float*)      d_out;     // [128*512*128] float32

    const int n_tok = in_sizes[0];              // 65536
    const unsigned grid = (unsigned)((n_tok + WAVES_PER_BLK - 1) / WAVES_PER_BLK);

    embedding_gather_kernel<<<grid, BLOCK_THREADS, 0, stream>>>(idx, emb, out, n_tok);
}